// GraphConvolutionNeuralNetwork_88373247082855
// MI455X (gfx1250) — compile-verified
//
#include <hip/hip_runtime.h>

typedef __attribute__((ext_vector_type(2))) float v2f;
typedef __attribute__((ext_vector_type(8))) float v8f;

#define N_NODES 50000
#define N_EDGES 800000
#define F_IN    9
#define F_HID   100
#define TILES_M (N_NODES / 16)          // 3125 (exact)
#define TILES_N ((F_HID + 15) / 16)     // 7

// ---------- degree / normalization ----------
__global__ void k_deg_init(float* __restrict__ deg) {
    int n = blockIdx.x * blockDim.x + threadIdx.x;
    if (n < N_NODES) deg[n] = 1.0f;     // self-loop contribution
}

__global__ void k_deg_count(const long long* __restrict__ ei, float* __restrict__ deg) {
    int e = blockIdx.x * blockDim.x + threadIdx.x;
    if (e < N_EDGES) atomicAdd(&deg[(int)ei[N_EDGES + e]], 1.0f);  // col = targets
}

__global__ void k_dis(const float* __restrict__ deg, float* __restrict__ dis) {
    int n = blockIdx.x * blockDim.x + threadIdx.x;
    if (n < N_NODES) dis[n] = rsqrtf(deg[n]);
}

// ---------- layer-1 GEMM via f32 WMMA: hs1 = (x @ W1) * dis[row]; acc1 = hs1 ----------
// Branchless staging: loads use clamped (always-valid) addresses, values are
// zero-selected afterwards -> no exec-mask branching around the WMMA.
__global__ void k_gemm1_wmma(const float* __restrict__ x, const float* __restrict__ W1,
                             const float* __restrict__ dis,
                             float* __restrict__ hs1, float* __restrict__ acc1) {
    const int lane = threadIdx.x & 31;
    const int gw = blockIdx.x * (blockDim.x >> 5) + (threadIdx.x >> 5);
    if (gw >= TILES_M * TILES_N) return;           // wave-uniform: EXEC stays all-1s
    const int tm = gw / TILES_N;
    const int tn = gw % TILES_N;
    const bool hi = lane >= 16;
    const int  l15 = lane & 15;
    const int  row = tm * 16 + l15;                // A-matrix row (always < N_NODES)
    const int  col = tn * 16 + l15;                // B/D column (>= F_HID on last tile pad)
    const bool colOk = (col < F_HID);
    const int  colC  = colOk ? col : (F_HID - 1); // clamped, always-valid address

    v8f c = {};
#pragma unroll
    for (int k0 = 0; k0 < 12; k0 += 4) {           // K padded 9 -> 12
        const int  ka  = k0 + (hi ? 2 : 0);
        const bool k0ok = (ka     < F_IN);
        const bool k1ok = (ka + 1 < F_IN);
        const int  ka0  = k0ok ? ka     : 0;       // clamped addresses
        const int  ka1  = k1ok ? ka + 1 : 0;

        const float ax = x[row * F_IN + ka0];
        const float ay = x[row * F_IN + ka1];
        const float bx = W1[ka0 * F_HID + colC];
        const float by = W1[ka1 * F_HID + colC];

        v2f a, b;
        a.x = k0ok ? ax : 0.0f;
        a.y = k1ok ? ay : 0.0f;
        b.x = (k0ok && colOk) ? bx : 0.0f;
        b.y = (k1ok && colOk) ? by : 0.0f;
        // 8 args: (neg_a, A, neg_b, B, c_mod, C, reuse_a, reuse_b)
        c = __builtin_amdgcn_wmma_f32_16x16x4_f32(false, a, false, b, (short)0, c,
                                                  false, false);
    }

    if (colOk) {
#pragma unroll
        for (int j = 0; j < 8; ++j) {
            const int m = tm * 16 + j + (hi ? 8 : 0);
            const float v = c[j] * dis[m];
            hs1[m * F_HID + col] = v;   // stable gather source
            acc1[m * F_HID + col] = v;  // accumulator seeded with self-loop term
        }
    }
}

// ---------- layer-1 edge scatter: acc1[col] += hs1[row] (one wave per edge) ----------
__global__ void k_scatter1(const long long* __restrict__ ei,
                           const float* __restrict__ hs1, float* __restrict__ acc1) {
    int t = blockIdx.x * blockDim.x + threadIdx.x;
    int e = t >> 5;
    int lane = t & 31;
    if (e >= N_EDGES) return;
    const int r = (int)ei[e];
    const int c = (int)ei[N_EDGES + e];
    const float* src = hs1 + (long)r * F_HID;
    float* dst = acc1 + (long)c * F_HID;
    for (int f = lane; f < F_HID; f += 32)
        atomicAdd(&dst[f], src[f]);
}

// ---------- layer-1 finish: h = relu(acc1 * dis[node] + b1) ----------
__global__ void k_relu(const float* __restrict__ acc1, const float* __restrict__ dis,
                       const float* __restrict__ b1, float* __restrict__ h) {
    int t = blockIdx.x * blockDim.x + threadIdx.x;
    if (t >= N_NODES * F_HID) return;
    int n = t / F_HID;
    int f = t - n * F_HID;
    h[t] = fmaxf(acc1[t] * dis[n] + b1[f], 0.0f);
}

// ---------- layer-2 linear: hs2[n] = dis[n] * dot(h[n,:], W2); acc2 = hs2 ----------
__global__ void k_lin2(const float* __restrict__ h, const float* __restrict__ W2,
                       const float* __restrict__ dis,
                       float* __restrict__ hs2, float* __restrict__ acc2) {
    int n = blockIdx.x * blockDim.x + threadIdx.x;
    if (n >= N_NODES) return;
    const float* hp = h + (long)n * F_HID;
    float s = 0.0f;
#pragma unroll 4
    for (int f = 0; f < F_HID; ++f) s += hp[f] * W2[f];
    s *= dis[n];
    hs2[n] = s;
    acc2[n] = s;   // self-loop seed
}

// ---------- layer-2 edge scatter ----------
__global__ void k_scatter2(const long long* __restrict__ ei,
                           const float* __restrict__ hs2, float* __restrict__ acc2) {
    int e = blockIdx.x * blockDim.x + threadIdx.x;
    if (e >= N_EDGES) return;
    atomicAdd(&acc2[(int)ei[N_EDGES + e]], hs2[(int)ei[e]]);
}

// ---------- output: out[n] = acc2[n] * dis[n] + b2 ----------
__global__ void k_out(const float* __restrict__ acc2, const float* __restrict__ dis,
                      const float* __restrict__ b2, float* __restrict__ out) {
    int n = blockIdx.x * blockDim.x + threadIdx.x;
    if (n < N_NODES) out[n] = acc2[n] * dis[n] + b2[0];
}

extern "C" void kernel_launch(void* const* d_in, const int* in_sizes, int n_in,
                              void* d_out, int out_size, void* d_ws, size_t ws_size,
                              hipStream_t stream) {
    const float*     x  = (const float*)d_in[0];
    const long long* ei = (const long long*)d_in[1];   // int64 [2, E]
    const float*     W1 = (const float*)d_in[2];
    const float*     b1 = (const float*)d_in[3];
    const float*     W2 = (const float*)d_in[4];
    const float*     b2 = (const float*)d_in[5];
    float*           out = (float*)d_out;

    // workspace layout (floats), ~41 MB total
    float* deg  = (float*)d_ws;                       // [N]
    float* dis  = deg  + N_NODES;                     // [N]
    float* hs1  = dis  + N_NODES;                     // [N,100]  (reused as h after scatter1)
    float* acc1 = hs1  + (size_t)N_NODES * F_HID;     // [N,100]
    float* hs2  = acc1 + (size_t)N_NODES * F_HID;     // [N]
    float* acc2 = hs2  + N_NODES;                     // [N]

    const int B = 256;

    k_deg_init<<<(N_NODES + B - 1) / B, B, 0, stream>>>(deg);
    k_deg_count<<<(N_EDGES + B - 1) / B, B, 0, stream>>>(ei, deg);
    k_dis<<<(N_NODES + B - 1) / B, B, 0, stream>>>(deg, dis);

    // layer 1: WMMA GEMM (one wave per 16x16 tile, 4 waves/block)
    {
        const int totalWaves = TILES_M * TILES_N;     // 21875
        const int wavesPerBlock = 4;
        k_gemm1_wmma<<<(totalWaves + wavesPerBlock - 1) / wavesPerBlock,
                       32 * wavesPerBlock, 0, stream>>>(x, W1, dis, hs1, acc1);
    }
    {
        const long totalThreads = (long)N_EDGES * 32;
        k_scatter1<<<(int)((totalThreads + B - 1) / B), B, 0, stream>>>(ei, hs1, acc1);
    }
    // h overwrites hs1 (hs1 no longer needed; stream order guarantees scatter1 is done)
    k_relu<<<(N_NODES * F_HID + B - 1) / B, B, 0, stream>>>(acc1, dis, b1, hs1);

    // layer 2
    k_lin2<<<(N_NODES + B - 1) / B, B, 0, stream>>>(hs1, W2, dis, hs2, acc2);
    k_scatter2<<<(N_EDGES + B - 1) / B, B, 0, stream>>>(ei, hs2, acc2);
    k_out<<<(N_NODES + B - 1) / B, B, 0, stream>>>(acc2, dis, b2, out);
}